// MaxMin_65721589564132
// MI455X (gfx1250) — compile-verified
//
#include <hip/hip_runtime.h>
#include <hip/hip_bf16.h>
#include <stdint.h>

// Problem sizes (fixed by the reference).
#define B_DIM 2048
#define K_DIM 512
#define N_DIM 512

// Tiling: 64x64 output tile per 256-thread block, K in chunks of 16,
// double-buffered in LDS via gfx1250 async global->LDS copies.
#define BM 64
#define BN 64
#define BK 16
#define NKT (K_DIM / BK)
#define NTHREADS 256

#if defined(__gfx1250__) &&                                                   \
    __has_builtin(__builtin_amdgcn_global_load_async_to_lds_b128) &&          \
    __has_builtin(__builtin_amdgcn_global_load_async_to_lds_b32)
#define USE_ASYNC 1
#else
#define USE_ASYNC 0
#endif

#if USE_ASYNC
// Builtin prototypes are typed: b128 takes (int4 AS1*, int4 AS3*, imm, imm),
// b32 takes (int AS1*, int AS3*, imm, imm).
typedef __attribute__((ext_vector_type(4))) int v4i;
typedef __attribute__((address_space(1))) v4i gv4i_t;
typedef __attribute__((address_space(3))) v4i lv4i_t;
typedef __attribute__((address_space(1))) int gi32_t;
typedef __attribute__((address_space(3))) int li32_t;

__device__ __forceinline__ void async_copy_b128(const float* gsrc, float* ldst) {
  __builtin_amdgcn_global_load_async_to_lds_b128((gv4i_t*)gsrc, (lv4i_t*)ldst, 0, 0);
}
__device__ __forceinline__ void async_copy_b32(const float* gsrc, float* ldst) {
  __builtin_amdgcn_global_load_async_to_lds_b32((gi32_t*)gsrc, (li32_t*)ldst, 0, 0);
}
__device__ __forceinline__ void wait_async_zero() {
#if __has_builtin(__builtin_amdgcn_s_wait_asynccnt)
  __builtin_amdgcn_s_wait_asynccnt(0);
#else
  asm volatile("s_wait_asynccnt 0" ::: "memory");
#endif
}
#endif  // USE_ASYNC

__global__ __launch_bounds__(NTHREADS) void MaxMin_65721589564132_kernel(
    const float* __restrict__ x,   // [B_DIM, K_DIM]
    const float* __restrict__ w,   // [K_DIM, N_DIM]
    float* __restrict__ out) {     // [B_DIM, N_DIM]
  // k-major LDS tiles so the per-k compute reads are contiguous b128 loads.
  __shared__ float xs[2][BK][BM];
  __shared__ float ws[2][BK][BN];

  const int tid = threadIdx.x;
  const int bn0 = blockIdx.x * BN;   // output-column tile base
  const int bm0 = blockIdx.y * BM;   // batch-row tile base

  // 16x16 thread grid; each thread owns a 4(m) x 4(n) accumulator tile.
  const int mt = (tid & 15) << 2;    // 0..60
  const int nt = (tid >> 4) << 2;    // 0..60

  float acc[4][4];
#pragma unroll
  for (int i = 0; i < 4; ++i)
#pragma unroll
    for (int j = 0; j < 4; ++j) acc[i][j] = -__builtin_inff();

  // ---- per-chunk compute (reads buffer bf) ----
  auto compute_chunk = [&](int bf) {
#pragma unroll
    for (int k = 0; k < BK; ++k) {
      const float4 av = *(const float4*)&xs[bf][k][mt];
      const float4 bv = *(const float4*)&ws[bf][k][nt];
      const float a[4] = {av.x, av.y, av.z, av.w};
      const float b[4] = {bv.x, bv.y, bv.z, bv.w};
#pragma unroll
      for (int i = 0; i < 4; ++i)
#pragma unroll
        for (int j = 0; j < 4; ++j)
          acc[i][j] = fmaxf(acc[i][j], fminf(a[i], b[j]));
    }
  };

#if USE_ASYNC
  // ---- async tile-load issue for chunk kt into buffer bf ----
  auto issue_chunk = [&](int kt, int bf) {
    const int k0 = kt * BK;
    // w tile: [BK][BN] contiguous 16B segments in both global and LDS.
    const int wk = tid >> 4;            // 0..15
    const int wn = (tid & 15) << 2;     // 0..60
    async_copy_b128(w + (size_t)(k0 + wk) * N_DIM + bn0 + wn, &ws[bf][wk][wn]);
    // x tile needs a 4B transpose (global k-fastest -> LDS m-fastest): b32 copies.
#pragma unroll
    for (int i = 0; i < 4; ++i) {
      const int e = tid + i * NTHREADS;  // 0..1023
      const int xk = e & (BK - 1);       // k fastest across lanes -> coalesced
      const int xm = e >> 4;             // 0..63
      async_copy_b32(x + (size_t)(bm0 + xm) * K_DIM + k0 + xk, &xs[bf][xk][xm]);
    }
  };

  issue_chunk(0, 0);
  for (int kt = 0; kt < NKT; ++kt) {
    const int bf = kt & 1;
    wait_async_zero();      // this wave's async copies (chunk kt) have landed
    __syncthreads();        // everyone's copies have landed; prev compute on bf^1 done
    if (kt + 1 < NKT) issue_chunk(kt + 1, bf ^ 1);
    compute_chunk(bf);
  }
#else
  // Fallback: synchronous loads through VGPRs (no CDNA5 async builtins available).
  for (int kt = 0; kt < NKT; ++kt) {
    __syncthreads();
    const int k0 = kt * BK;
    const int wk = tid >> 4;
    const int wn = (tid & 15) << 2;
    *(float4*)&ws[0][wk][wn] =
        *(const float4*)(w + (size_t)(k0 + wk) * N_DIM + bn0 + wn);
#pragma unroll
    for (int i = 0; i < 4; ++i) {
      const int e = tid + i * NTHREADS;
      const int xk = e & (BK - 1);
      const int xm = e >> 4;
      xs[0][xk][xm] = x[(size_t)(bm0 + xm) * K_DIM + k0 + xk];
    }
    __syncthreads();
    compute_chunk(0);
  }
#endif

  // ---- store 4x4 tile as four float4 rows ----
#pragma unroll
  for (int i = 0; i < 4; ++i) {
    float4 v = make_float4(acc[i][0], acc[i][1], acc[i][2], acc[i][3]);
    *(float4*)(out + (size_t)(bm0 + mt + i) * N_DIM + bn0 + nt) = v;
  }
}

extern "C" void kernel_launch(void* const* d_in, const int* in_sizes, int n_in,
                              void* d_out, int out_size, void* d_ws, size_t ws_size,
                              hipStream_t stream) {
  (void)in_sizes; (void)n_in; (void)out_size; (void)d_ws; (void)ws_size;
  const float* x = (const float*)d_in[0];   // [2048, 512]
  const float* w = (const float*)d_in[1];   // [512, 512]
  float* out = (float*)d_out;               // [2048, 512]

  dim3 grid(N_DIM / BN, B_DIM / BM);        // (8, 32) = 256 blocks
  dim3 block(NTHREADS);
  MaxMin_65721589564132_kernel<<<grid, block, 0, stream>>>(x, w, out);
}